// DepthwiseCausalConv_15109694948066
// MI455X (gfx1250) — compile-verified
//
#include <hip/hip_runtime.h>
#include <stdint.h>

// Problem constants from the reference (B=8, T=4096, C=512, K=4, fp32).
#define BB 8
#define TD 4096
#define CD 512
#define TT 64      // time steps per block tile
#define NSTAGE 16  // async pipeline depth (LDS ring slots)

typedef __attribute__((ext_vector_type(4))) float f4;

// Depthwise causal conv1d, streaming over time.
// Block = 128 threads: lane i owns channels 4i..4i+3 (128*4 = 512 = C).
// Each block handles one (batch, 64-step time tile).
// Data path: global_load_async_to_lds_b128 (ASYNCcnt-tracked, 16 rows in
// flight per wave) -> ds_load_b128 -> rolling f4 register window -> FMA ->
// nontemporal global_store_b128.
__global__ __launch_bounds__(128) void dwconv_causal_kernel(
    const float* __restrict__ x, const float* __restrict__ w,
    float* __restrict__ out)
{
    __shared__ alignas(16) float lds[NSTAGE * CD];

    const int tid  = threadIdx.x;       // 0..127
    const int c0   = tid * 4;           // first of this lane's 4 channels
    const int tilesPerB = TD / TT;      // 64
    const int b    = blockIdx.x / tilesPerB;
    const int tile = blockIdx.x % tilesPerB;
    const int t0   = tile * TT;

    // Per-channel tap vectors: weight is [C,4] row-major, so the 4 taps of
    // channel c are contiguous -> one 16B load per channel.
    const f4 W0 = *(const f4*)(w + (size_t)(c0 + 0) * 4);
    const f4 W1 = *(const f4*)(w + (size_t)(c0 + 1) * 4);
    const f4 W2 = *(const f4*)(w + (size_t)(c0 + 2) * 4);
    const f4 W3 = *(const f4*)(w + (size_t)(c0 + 3) * 4);

    const float* xb = x + (size_t)b * TD * CD;
    // addrspace(3) -> flat cast keeps the LDS byte offset in the low 32 bits,
    // which is exactly what the async op's LDS-address VGPR wants.
    const uint32_t lds_base = (uint32_t)(uintptr_t)(&lds[0]);

    // ---- prologue: put rows 0..NSTAGE-1 in flight (row r <-> time t0-3+r) --
    #pragma unroll
    for (int r = 0; r < NSTAGE; ++r) {
        int t = t0 - 3 + r;
        int tc = t < 0 ? 0 : t;  // clamp; value zeroed at consume for t<0
        const float* gp = xb + (size_t)tc * CD + c0;
        uint32_t la = lds_base + (uint32_t)((r * CD + c0) * 4);
        asm volatile("global_load_async_to_lds_b128 %0, %1, off"
                     :: "v"(la), "v"(gp) : "memory");
    }
    // 16 issued; <=13 outstanding => rows 0..2 have landed (in-order returns).
    asm volatile("s_wait_asynccnt 13" ::: "memory");

    f4 h0 = *(const f4*)&lds[0 * CD + c0];  // time t0-3  (tap 0 history)
    f4 h1 = *(const f4*)&lds[1 * CD + c0];  // time t0-2
    f4 h2 = *(const f4*)&lds[2 * CD + c0];  // time t0-1
    if (t0 == 0) {  // causal left padding: history before t=0 is zero
        const f4 z = {0.f, 0.f, 0.f, 0.f};
        h0 = z; h1 = z; h2 = z;
    }

    // ---- steady state: one async issue + one wait + one row consumed ------
    for (int i = 0; i < TT; ++i) {
        // Issue row i+NSTAGE (slot (i)&15 is free: row i was consumed at
        // iteration i-3). Tail issues clamp to the last real row (TT+2):
        // they re-read an L2-resident line so the wait immediate stays const.
        int r  = i + NSTAGE;
        int rr = r > TT + 2 ? TT + 2 : r;
        const float* gp = xb + (size_t)(t0 - 3 + rr) * CD + c0;
        uint32_t la = lds_base + (uint32_t)(((r & (NSTAGE - 1)) * CD + c0) * 4);
        asm volatile("global_load_async_to_lds_b128 %0, %1, off"
                     :: "v"(la), "v"(gp) : "memory");

        // issued = i+17; <=12 outstanding => rows 0..i+4 complete (need i+3).
        asm volatile("s_wait_asynccnt 12" ::: "memory");

        const f4 v = *(const f4*)&lds[((i + 3) & (NSTAGE - 1)) * CD + c0];

        f4 o;
        o.x = W0.x * h0.x + W0.y * h1.x + W0.z * h2.x + W0.w * v.x;
        o.y = W1.x * h0.y + W1.y * h1.y + W1.z * h2.y + W1.w * v.y;
        o.z = W2.x * h0.z + W2.y * h1.z + W2.z * h2.z + W2.w * v.z;
        o.w = W3.x * h0.w + W3.y * h1.w + W3.z * h2.w + W3.w * v.w;

        __builtin_nontemporal_store(
            o, (f4*)(out + ((size_t)b * TD + t0 + i) * CD + c0));

        h0 = h1; h1 = h2; h2 = v;  // roll the 4-tap window
    }
}

extern "C" void kernel_launch(void* const* d_in, const int* in_sizes, int n_in,
                              void* d_out, int out_size, void* d_ws, size_t ws_size,
                              hipStream_t stream) {
    const float* x = (const float*)d_in[0];   // [B,T,C] fp32
    const float* w = (const float*)d_in[1];   // [C,K]   fp32
    float* out = (float*)d_out;               // [B,T,C] fp32
    (void)in_sizes; (void)n_in; (void)out_size; (void)d_ws; (void)ws_size;

    const int blocks = BB * (TD / TT);        // 8 * 64 = 512 blocks
    hipLaunchKernelGGL(dwconv_causal_kernel, dim3(blocks), dim3(128), 0, stream,
                       x, w, out);
}